// TransformerEncoder_23570780520616
// MI455X (gfx1250) — compile-verified
//
#include <hip/hip_runtime.h>
#include <math.h>

typedef float v2f __attribute__((ext_vector_type(2)));
typedef float v8f __attribute__((ext_vector_type(8)));

#define B_ 16
#define S_ 197
#define E_ 768
#define M_ 12
#define H_ 64
#define TOK 3152          // B_*S_
#define NELEM 2420736L    // B_*S_*E_

// ---- FP32 WMMA 16x16x4:  D = A(16x4) * B(4x16) + C ----
__device__ __forceinline__ v8f wmma4(v2f a, v2f b, v8f c) {
  return __builtin_amdgcn_wmma_f32_16x16x4_f32(false, a, false, b, (short)0, c,
                                               false, false);
}

// ---- CDNA5 async global->LDS copy (tracked by ASYNCcnt) ----
// Generic pointers to LDS carry the byte offset in their low 32 bits.
__device__ __forceinline__ void async_b128(void* lds, const void* g) {
  unsigned loff = (unsigned)(unsigned long long)lds;
  unsigned long long ga = (unsigned long long)g;
  asm volatile("global_load_async_to_lds_b128 %0, %1, off" ::"v"(loff),
               "v"(ga)
               : "memory");
}
__device__ __forceinline__ void wait_async0() {
  asm volatile("s_wait_asynccnt 0x0" ::: "memory");
}
__device__ __forceinline__ void wait_async2() {
  asm volatile("s_wait_asynccnt 0x2" ::: "memory");
}

// =====================================================================
// Generic 64x64-tile WMMA GEMM, 256 threads = 8 waves (4M x 2N),
// each wave owns a 16x32 strip (two 16x16 C frags). BK = 16.
// Double-buffered LDS tiles fed by async b128 copies.
// mode 0: QKV proj   A=x(3152x768), B=W[z](768x768), scatter to (B,M,L,S,H)
// mode 1: out proj   A=ctx rows for module z,        C -> mha concat
// mode 2: MLP1       plain row-major, epilogue = exact GELU
// mode 3: MLP2       plain row-major, epilogue = + aux (residual y1)
// =====================================================================
__global__ __launch_bounds__(256) void gemm64(
    int mode, const float* __restrict__ A, const float* __restrict__ Bmat,
    float* __restrict__ C, const float* __restrict__ aux, int Mrows, int Kdim,
    int ldb, long zBstride) {
  __shared__ float As[2][64 * 20];  // stride 20 floats = 80B (16B aligned)
  __shared__ float Bs[2][16 * 80];  // stride 80 floats = 320B

  const int tid = threadIdx.x;
  const int z = blockIdx.z;
  const int tileM = blockIdx.x * 64;
  const int tileN = blockIdx.y * 64;
  const float* Bz = Bmat + (long)z * zBstride;

  const int wave = tid >> 5, lane = tid & 31;
  const int half = lane >> 4, lid = lane & 15;
  const int wm = wave >> 1, wn = wave & 1;

  // A-tile loader coords: 64 rows x 16 cols, one b128/thread/chunk.
  // Rows past Mrows are clamped (same data as last row): their C rows are
  // never stored, and this keeps per-wave async counts uniform (no EXEC
  // divergence around the async issue).
  const int arow = tid >> 2;
  const int ac4 = (tid & 3) << 2;
  const int grA = tileM + arow;
  const int grAc = grA < Mrows ? grA : (Mrows - 1);
  long aoff;
  if (mode == 1) {
    int bb = grAc / S_, ss = grAc - bb * S_;
    aoff = ((long)(bb * M_ + z) * S_ + ss) * (long)Kdim;
  } else {
    aoff = (long)grAc * (long)Kdim;
  }
  const float* Ap = A + aoff + ac4;
  // B-tile loader coords: 16 rows x 64 cols, one b128/thread/chunk
  const int brow = tid >> 4;
  const int bc4 = (tid & 15) << 2;
  const float* Bp = Bz + (long)brow * ldb + tileN + bc4;

  v8f c0 = {};
  v8f c1 = {};

  const int nch = Kdim >> 4;
  // prologue: chunk 0 -> buffer 0
  async_b128(&As[0][arow * 20 + ac4], Ap);
  async_b128(&Bs[0][brow * 80 + bc4], Bp);

  for (int ch = 0; ch < nch; ++ch) {
    const float* as = As[ch & 1];
    const float* bs = Bs[ch & 1];
    if (ch + 1 < nch) {  // issue next chunk into the other buffer
      async_b128(&As[(ch + 1) & 1][arow * 20 + ac4], Ap + (ch + 1) * 16);
      async_b128(&Bs[(ch + 1) & 1][brow * 80 + bc4],
                 Bp + (long)(ch + 1) * 16 * ldb);
      wait_async2();  // oldest 2 (current chunk) have landed
    } else {
      wait_async0();
    }
    __syncthreads();
#pragma unroll
    for (int k4 = 0; k4 < 4; ++k4) {
      const int kb = k4 * 4 + 2 * half;
      v2f a = *(const v2f*)(&as[(wm * 16 + lid) * 20 + kb]);
      v2f b0, b1;
      b0.x = bs[kb * 80 + wn * 32 + lid];
      b0.y = bs[(kb + 1) * 80 + wn * 32 + lid];
      b1.x = bs[kb * 80 + wn * 32 + 16 + lid];
      b1.y = bs[(kb + 1) * 80 + wn * 32 + 16 + lid];
      c0 = wmma4(a, b0, c0);
      c1 = wmma4(a, b1, c1);
    }
    __syncthreads();  // everyone done with this buffer before it is refilled
  }

#pragma unroll
  for (int f = 0; f < 2; ++f) {
    v8f cc = f ? c1 : c0;
#pragma unroll
    for (int r = 0; r < 8; ++r) {
      const int gr = tileM + wm * 16 + r + 8 * half;
      const int gc = tileN + wn * 32 + f * 16 + lid;
      if (gr < Mrows) {
        float vv = cc[r];
        if (mode == 0) {
          int bb = gr / S_, ss = gr - bb * S_;
          int ll = gc >> 6, hh = gc & 63;
          C[(((long)(bb * M_ + z) * M_ + ll) * S_ + ss) * H_ + hh] = vv;
        } else if (mode == 1) {
          C[(long)gr * E_ + z * H_ + gc] = vv;
        } else if (mode == 2) {
          C[(long)gr * ldb + gc] =
              0.5f * vv * (1.0f + erff(vv * 0.70710678118654752f));
        } else {
          C[(long)gr * ldb + gc] = vv + aux[(long)gr * ldb + gc];
        }
      }
    }
  }
}

// =====================================================================
// Attention: one block per (b, module, inner-head). K,V resident in LDS
// (320KB/WGP), streamed in with async b128 copies. 2 waves; each wave owns
// a 16-query tile per iteration: scores (WMMA) -> softmax -> ctx (WMMA).
// =====================================================================
#define KSTR 68   // 272B rows: 16B aligned, 16 frag rows -> distinct banks
#define VSTR 80   // 320B rows: half-waves 32 banks apart
#define SSTR 210  // score scratch stride
#define SPAD 208  // 13 * 16 (keys padded)

__global__ __launch_bounds__(64) void attn_kernel(const float* __restrict__ q,
                                                  const float* __restrict__ k,
                                                  const float* __restrict__ v,
                                                  float* __restrict__ ctx) {
  extern __shared__ float lds[];
  float* Klds = lds;                 // 208 x 68
  float* Vlds = lds + SPAD * KSTR;   // 208 x 80
  float* Slds = Vlds + SPAD * VSTR;  // 2 waves x 16 x 210

  const int bml = blockIdx.x;
  const int ll = bml % M_;
  const int md = (bml / M_) % M_;
  const int bb = bml / (M_ * M_);
  const long base = (long)bml * S_ * H_;
  const float* Qg = q + base;
  const float* Kg = k + base;
  const float* Vg = v + base;

  const int tid = threadIdx.x;
  // zero the pad rows (async copies cannot zero-fill)
  for (int r = S_; r < SPAD; ++r) {
    for (int c = tid; c < KSTR; c += 64) Klds[r * KSTR + c] = 0.f;
    for (int c = tid; c < VSTR; c += 64) Vlds[r * VSTR + c] = 0.f;
  }
  // stream K and V into LDS: 16B segments, async
  for (int i = tid; i < S_ * 16; i += 64) {
    int row = i >> 4, seg = (i & 15) << 2;
    async_b128(&Klds[row * KSTR + seg], Kg + row * H_ + seg);
    async_b128(&Vlds[row * VSTR + seg], Vg + row * H_ + seg);
  }
  wait_async0();
  __syncthreads();

  const int wave = tid >> 5, lane = tid & 31;
  const int half = lane >> 4, lid = lane & 15;
  float* Sw = Slds + wave * 16 * SSTR;

  for (int qt0 = 0; qt0 < 13; qt0 += 2) {
    const int qt = qt0 + wave;
    const bool active = qt < 13;  // wave-uniform => EXEC stays full for WMMA
    const int qbase = qt * 16;
    if (active) {
      int qrow = qbase + lid;
      if (qrow > S_ - 1) qrow = S_ - 1;  // clamp: garbage rows never stored
      for (int nt = 0; nt < 13; ++nt) {
        v8f c = {};
#pragma unroll
        for (int k4 = 0; k4 < 16; ++k4) {
          const int kb = k4 * 4 + 2 * half;
          v2f a = *(const v2f*)(Qg + (long)qrow * H_ + kb);
          v2f b;
          b.x = Klds[(nt * 16 + lid) * KSTR + kb];
          b.y = Klds[(nt * 16 + lid) * KSTR + kb + 1];
          c = wmma4(a, b, c);
        }
#pragma unroll
        for (int r = 0; r < 8; ++r)
          Sw[(r + 8 * half) * SSTR + nt * 16 + lid] = c[r] * 0.125f;
      }
    }
    __syncthreads();
    if (active && lane < 16) {  // softmax: one lane per query row
      float* Srow = Sw + lane * SSTR;
      if (qbase + lane < S_) {
        float mx = -3.0e38f;
        for (int j = 0; j < S_; ++j) mx = fmaxf(mx, Srow[j]);
        float sum = 0.f;
        for (int j = 0; j < S_; ++j) {
          float e = __expf(Srow[j] - mx);
          Srow[j] = e;
          sum += e;
        }
        float inv = 1.0f / sum;
        for (int j = 0; j < S_; ++j) Srow[j] *= inv;
        for (int j = S_; j < SPAD; ++j) Srow[j] = 0.f;
      } else {
        for (int j = 0; j < SPAD; ++j) Srow[j] = 0.f;
      }
    }
    __syncthreads();
    if (active) {
      for (int nt = 0; nt < 4; ++nt) {
        v8f c = {};
        for (int k4 = 0; k4 < 52; ++k4) {
          const int kb = k4 * 4 + 2 * half;
          v2f a = *(const v2f*)(&Sw[lid * SSTR + kb]);
          v2f b;
          b.x = Vlds[kb * VSTR + nt * 16 + lid];
          b.y = Vlds[(kb + 1) * VSTR + nt * 16 + lid];
          c = wmma4(a, b, c);
        }
#pragma unroll
        for (int r = 0; r < 8; ++r) {
          int qq = qbase + r + 8 * half;
          if (qq < S_) {
            ctx[((long)(bb * M_ + md) * S_ + qq) * E_ + ll * H_ + nt * 16 +
                lid] = c[r];
          }
        }
      }
    }
    __syncthreads();
  }
}

// ---- whole-tensor sum / sum-of-squares in double (f64 global atomics) ----
__global__ __launch_bounds__(256) void reduce_sums(const float* __restrict__ p,
                                                   long n,
                                                   double* __restrict__ acc) {
  __shared__ double s1[256], s2[256];
  double a = 0.0, b = 0.0;
  for (long i = blockIdx.x * 256L + threadIdx.x; i < n;
       i += (long)gridDim.x * 256L) {
    double x = (double)p[i];
    a += x;
    b += x * x;
  }
  s1[threadIdx.x] = a;
  s2[threadIdx.x] = b;
  __syncthreads();
  for (int off = 128; off > 0; off >>= 1) {
    if ((int)threadIdx.x < off) {
      s1[threadIdx.x] += s1[threadIdx.x + off];
      s2[threadIdx.x] += s2[threadIdx.x + off];
    }
    __syncthreads();
  }
  if (threadIdx.x == 0) {
    atomicAdd(acc, s1[0]);
    atomicAdd(acc + 1, s2[0]);
  }
}

// ---- y = (addx ? x : 0) + (in - mu)*rsqrt(var+eps)*g + b ----
__global__ __launch_bounds__(256) void ln_ew_kernel(
    const float* __restrict__ x, const float* __restrict__ in,
    const float* __restrict__ g, const float* __restrict__ bta,
    const double* __restrict__ acc, long n, float* __restrict__ y, int addx) {
  const double mu = acc[0] / (double)n;
  const double var = acc[1] / (double)n - mu * mu;
  const float rs = (float)(1.0 / sqrt(var + 1e-5));
  const float muf = (float)mu;
  for (long i = blockIdx.x * (long)blockDim.x + threadIdx.x; i < n;
       i += (long)gridDim.x * blockDim.x) {
    float lnv = (in[i] - muf) * rs * g[i] + bta[i];
    y[i] = addx ? (x[i] + lnv) : lnv;
  }
}

extern "C" void kernel_launch(void* const* d_in, const int* in_sizes, int n_in,
                              void* d_out, int out_size, void* d_ws,
                              size_t ws_size, hipStream_t stream) {
  const float* x = (const float*)d_in[0];
  const float* Wq = (const float*)d_in[1];
  const float* Wk = (const float*)d_in[2];
  const float* Wv = (const float*)d_in[3];
  const float* Wo = (const float*)d_in[4];
  const float* W1 = (const float*)d_in[5];
  const float* W2 = (const float*)d_in[6];
  const float* ln1g = (const float*)d_in[7];
  const float* ln1b = (const float*)d_in[8];
  const float* ln2g = (const float*)d_in[9];
  const float* ln2b = (const float*)d_in[10];
  float* out = (float*)d_out;

  char* ws = (char*)d_ws;
  double* acc = (double*)ws;  // 4 doubles
  const long QKVN = (long)B_ * M_ * M_ * S_ * H_;  // 29,048,832 floats
  float* qb = (float*)(ws + 256);
  float* kb = qb + QKVN;
  float* vb = kb + QKVN;
  float* ctx = vb + QKVN;
  float* mha = ctx + QKVN;
  float* y1 = mha + NELEM;
  float* h = y1 + NELEM;
  float* t = h + NELEM;

  hipMemsetAsync(acc, 0, 4 * sizeof(double), stream);

  // QKV projections: 12 modules in grid.z, one launch per weight tensor
  gemm64<<<dim3(50, 12, 12), 256, 0, stream>>>(0, x, Wq, qb, nullptr, TOK, E_,
                                               E_, (long)E_ * E_);
  gemm64<<<dim3(50, 12, 12), 256, 0, stream>>>(0, x, Wk, kb, nullptr, TOK, E_,
                                               E_, (long)E_ * E_);
  gemm64<<<dim3(50, 12, 12), 256, 0, stream>>>(0, x, Wv, vb, nullptr, TOK, E_,
                                               E_, (long)E_ * E_);

  // Attention per (b, module, inner head): 150,016 B dynamic LDS
  attn_kernel<<<dim3(B_ * M_ * M_), 64,
                (SPAD * KSTR + SPAD * VSTR + 2 * 16 * SSTR) * 4, stream>>>(
      qb, kb, vb, ctx);

  // Out projection + module-major concat -> mha (B,S,E)
  gemm64<<<dim3(50, 1, 12), 256, 0, stream>>>(1, ctx, Wo, mha, nullptr, TOK,
                                              E_, H_, (long)E_ * H_);

  // y1 = x + full_layernorm(mha)
  reduce_sums<<<512, 256, 0, stream>>>(mha, NELEM, acc);
  ln_ew_kernel<<<1024, 256, 0, stream>>>(x, mha, ln1g, ln1b, acc, NELEM, y1, 1);

  // h = full_layernorm(y1)
  reduce_sums<<<512, 256, 0, stream>>>(y1, NELEM, acc + 2);
  ln_ew_kernel<<<1024, 256, 0, stream>>>(y1, y1, ln2g, ln2b, acc + 2, NELEM, h,
                                         0);

  // MLP: t = gelu(h @ W1);  out = y1 + t @ W2
  gemm64<<<dim3(50, 24, 1), 256, 0, stream>>>(2, h, W1, t, nullptr, TOK, E_,
                                              2 * E_, 0);
  gemm64<<<dim3(50, 12, 1), 256, 0, stream>>>(3, t, W2, out, y1, TOK, 2 * E_,
                                              E_, 0);
}